// UniformQuantizer_22557168239298
// MI455X (gfx1250) — compile-verified
//
#include <hip/hip_runtime.h>

// Elementwise log-sqrt2 quantizer, MI455X (gfx1250, wave32).
// Memory-bound: ~902 MB traffic/call -> ~39 us floor at 23.3 TB/s.
// v2: 2D grid (row = blockIdx.y) so per-row scale/asym become block-uniform
//     scalar (SMEM) loads and the integer row-division disappears; one
//     hoisted IEEE reciprocal per thread instead of 4 IEEE divides; copysign
//     instead of explicit sign select (x==0 always lands in the zero_flag
//     path, since log2(1e-32) clamps to `lower`).

typedef float v4f __attribute__((ext_vector_type(4)));

#define Q_ROWS   4096
#define Q_COLS   11008
#define Q_COLS4  (Q_COLS / 4)           // 2752 float4 per row (exact)
#define Q_BLK    256                    // 8 wave32 per block
#define Q_XBLKS  ((Q_COLS4 + Q_BLK - 1) / Q_BLK)   // 11 blocks per row

#define SQRT2_F        1.41421356237309504880f   // np.float32(sqrt(2)) == 0x3FB504F3
#define APPROX_FACTOR  1.06066017177982128660f   // np.float32(1.5/sqrt(2))

__global__ __launch_bounds__(Q_BLK) void logquant_kernel(
    const float* __restrict__ x,
    const float* __restrict__ scale,     // [ROWS,1]
    const float* __restrict__ zero,
    const float* __restrict__ code,      // exactly 2.0 or sqrt(2)
    const float* __restrict__ level,
    const float* __restrict__ asym,      // [ROWS,1], 0 or 1
    float* __restrict__ out)
{
    const int col4 = blockIdx.x * Q_BLK + threadIdx.x;
    if (col4 >= Q_COLS4) return;
    const int row = blockIdx.y;
    const int i   = row * Q_COLS4 + col4;

    // Block-uniform row scalars -> scalar SMEM loads (stay L2-resident, RT).
    const float s     = scale[row];
    const float a     = asym[row];
    const float lower = -1.0f - 0.5f * a;
    const float inv_s = 1.0f / s;        // one IEEE div per thread (uniform)

    // Big streams: non-temporal 128-bit loads (no reuse; 902 MB >> 192 MB L2).
    const v4f xv = __builtin_nontemporal_load((const v4f*)x     + i);
    const v4f zv = __builtin_nontemporal_load((const v4f*)zero  + i);
    const v4f cv = __builtin_nontemporal_load((const v4f*)code  + i);
    const v4f lv = __builtin_nontemporal_load((const v4f*)level + i);

    v4f ov;
#pragma unroll
    for (int k = 0; k < 4; ++k) {
        const float xx = xv[k];
        const float z  = zv[k];
        const float c  = cv[k];
        const float l  = lv[k];

        // t = |x/scale| + 1e-32   (>= 1e-32, always normal -> raw v_log_f32 ok)
        const float t = fabsf(xx * inv_s) + 1e-32f;

        // ln(t)/ln(c): c is exactly 2 or sqrt(2) -> log2(t) * {1,2}
        const bool  is_s2 = (c == SQRT2_F);
        const float xlog  = __builtin_amdgcn_logf(t) * (is_s2 ? 2.0f : 1.0f);

        // jnp.round = round-half-even -> v_rndne_f32 (default RNE mode)
        const float xint = rintf(xlog);

        const float xc1 = fmaxf(xint - z, lower);       // clamp(min=lower)
        const float xc  = fminf(xc1 - l, -1.0f);        // clamp(max=-1)
        const float e   = xc + l + z;                   // final exponent

        // pow(c, e) = exp2(e * log2(c)) -> single v_exp_f32
        float q = __builtin_amdgcn_exp2f(e * (is_s2 ? 0.5f : 1.0f));

        // sqrt2 odd-exponent hardware approximation (e is integer-valued
        // whenever the result survives; half-integer e only occurs when the
        // zero_flag path below forces 0)
        const int ei = (int)e;
        if (is_s2 && (ei & 1)) q *= APPROX_FACTOR;

        // q*s >= 0, so sign(x)*|q*s| == copysign(q*s, x); the x==0 case is
        // guaranteed zero_flag (log2(1e-32) ~ -106 clamps to `lower`).
        const float v = __builtin_copysignf(q * s, xx);

        // underflow -> exact zero
        ov[k] = (xc1 <= lower) ? 0.0f : v;
    }

    // Streaming 128-bit NT store (write-once data; don't pollute L2).
    __builtin_nontemporal_store(ov, (v4f*)out + i);
}

extern "C" void kernel_launch(void* const* d_in, const int* in_sizes, int n_in,
                              void* d_out, int out_size, void* d_ws, size_t ws_size,
                              hipStream_t stream) {
    (void)in_sizes; (void)n_in; (void)out_size; (void)d_ws; (void)ws_size;

    const float* x     = (const float*)d_in[0];
    const float* scale = (const float*)d_in[1];
    const float* zero  = (const float*)d_in[2];
    const float* code  = (const float*)d_in[3];
    const float* level = (const float*)d_in[4];
    const float* asym  = (const float*)d_in[5];
    float*       out   = (float*)d_out;

    dim3 grid(Q_XBLKS, Q_ROWS);          // 11 x 4096 blocks
    logquant_kernel<<<grid, Q_BLK, 0, stream>>>(x, scale, zero, code,
                                                level, asym, out);
}